// RandomRoi_29652454211707
// MI455X (gfx1250) — compile-verified
//
#include <hip/hip_runtime.h>
#include <stdint.h>

#define NMS_THR   0.7f
#define MIN_SCALE 0.1f
#define MAX_FINAL 50
#define SORT_N    2048          // next pow2 >= N_FIRST (2000)
#define BLOCK     256

typedef unsigned int u32x4 __attribute__((ext_vector_type(4)));
typedef int          i32x4 __attribute__((ext_vector_type(4)));
typedef int          i32x8 __attribute__((ext_vector_type(8)));

// Low 32 bits of a generic pointer into LDS == LDS byte offset
__device__ __forceinline__ uint32_t lds_byte_off(const void* p) {
    return (uint32_t)(uintptr_t)p;
}

// Issue one TDM descriptor: 1-D tensor of `nelem` 4-byte elements,
// global -> LDS. D# layout per CDNA5 ISA ch. 7/8 (async_tensor doc).
__device__ __forceinline__ void tdm_load_1d(uint32_t lds_off, uint64_t gaddr,
                                            uint32_t nelem) {
#if __has_builtin(__builtin_amdgcn_tensor_load_to_lds)
    u32x4 g0;
    g0[0] = 0x1u;                                   // count=1, user mode, no gather
    g0[1] = lds_off;                                // lds_addr (bytes)
    g0[2] = (uint32_t)(gaddr & 0xFFFFFFFFu);        // global_addr[31:0]
    g0[3] = (uint32_t)((gaddr >> 32) & 0x1FFFFFFu)  // global_addr[56:32]
          | (2u << 30);                             // type = 2 ("image")

    i32x8 g1;
    g1[0] = (int)(2u << 16);                              // wg_mask=0, data_size=4B
    g1[1] = (int)((nelem & 0xFFFFu) << 16);               // tensor_dim0[15:0]  @63:48
    g1[2] = (int)(((nelem >> 16) & 0xFFFFu) | (1u << 16));// dim0[31:16] | tensor_dim1=1
    g1[3] = (int)((nelem & 0xFFFFu) << 16);               // tile_dim0 @127:112
    g1[4] = 0;                                            // tile_dim1=0, tile_dim2=0
    g1[5] = (int)nelem;                                   // tensor_dim0_stride[31:0]
    g1[6] = 0;                                            // stride0[47:32], stride1[15:0]
    g1[7] = 0;                                            // stride1[47:16]

    i32x4 gz = {0, 0, 0, 0};
#if defined(__clang_major__) && (__clang_major__ >= 23)
    i32x8 gz8 = {0, 0, 0, 0, 0, 0, 0, 0};
    __builtin_amdgcn_tensor_load_to_lds(g0, g1, gz, gz, gz8, 0);
#else
    __builtin_amdgcn_tensor_load_to_lds(g0, g1, gz, gz, 0);
#endif
#else
    (void)lds_off; (void)gaddr; (void)nelem;   // fallback handled by caller flag
#endif
}

__global__ void __launch_bounds__(BLOCK)
roi_nms_kernel(const float* __restrict__ im_info,
               const float* __restrict__ boxes,
               const float* __restrict__ scores,
               const int*   __restrict__ counts,
               float*       __restrict__ out,
               int B, int N)
{
    __shared__ float sbx[SORT_N * 4];       // scaled boxes (x1,y1,x2,y2)
    __shared__ float ssc[SORT_N];           // raw scores
    __shared__ float skey[SORT_N];          // sort keys (eff score)
    __shared__ int   sval[SORT_N];          // sort payload (box index)
    __shared__ float skept[MAX_FINAL * 4];  // kept boxes in keep order
    __shared__ int   s_nk;

    const int b   = blockIdx.x;
    const int tid = threadIdx.x;

    // ---- Stage raw boxes + scores into LDS via the Tensor Data Mover ----
#if __has_builtin(__builtin_amdgcn_tensor_load_to_lds)
    if (tid < 32) {   // one wave issues both DMA descriptors
        tdm_load_1d(lds_byte_off(sbx),
                    (uint64_t)(uintptr_t)(boxes + (size_t)b * N * 4),
                    (uint32_t)(N * 4));
        tdm_load_1d(lds_byte_off(ssc),
                    (uint64_t)(uintptr_t)(scores + (size_t)b * N),
                    (uint32_t)N);
        __builtin_amdgcn_s_wait_tensorcnt(0);
    }
#else
    for (int i = tid; i < N * 4; i += BLOCK) sbx[i] = boxes[(size_t)b * N * 4 + i];
    for (int i = tid; i < N;     i += BLOCK) ssc[i] = scores[(size_t)b * N + i];
#endif
    __syncthreads();

    const float H = im_info[b * 3 + 0];
    const float W = im_info[b * 3 + 1];
    const float NEG_INF = -__builtin_inff();

    // ---- Normalize + scale + size-filter; build sort keys ----
    for (int i = tid; i < SORT_N; i += BLOCK) {
        float key = NEG_INF;
        if (i < N) {
            float bx0 = sbx[i * 4 + 0], by0 = sbx[i * 4 + 1];
            float bx1 = sbx[i * 4 + 2], by1 = sbx[i * 4 + 3];
            float x1 = fminf(bx0, bx1) * W;
            float x2 = fmaxf(bx0, bx1) * W;
            float y1 = fminf(by0, by1) * H;
            float y2 = fmaxf(by0, by1) * H;
            sbx[i * 4 + 0] = x1; sbx[i * 4 + 1] = y1;   // own-slot rewrite: no race
            sbx[i * 4 + 2] = x2; sbx[i * 4 + 3] = y2;
            bool valid = ((y2 - y1) > MIN_SCALE * H) && ((x2 - x1) > MIN_SCALE * W);
            key = valid ? ssc[i] : NEG_INF;
        }
        skey[i] = key;
        sval[i] = i;
    }
    __syncthreads();

    // ---- Bitonic sort, descending by key ----
    for (int k = 2; k <= SORT_N; k <<= 1) {
        for (int j = k >> 1; j > 0; j >>= 1) {
            for (int i = tid; i < SORT_N; i += BLOCK) {
                int p = i ^ j;
                if (p > i) {
                    float a = skey[i], c = skey[p];
                    bool sw = ((i & k) == 0) ? (a < c) : (a > c);
                    if (sw) {
                        skey[i] = c; skey[p] = a;
                        int t = sval[i]; sval[i] = sval[p]; sval[p] = t;
                    }
                }
            }
            __syncthreads();
        }
    }

    // ---- Greedy NMS: wave 0 scans sorted list, kept set lives in lane regs ----
    // Greedy identity: box kept <=> overlaps no earlier *kept* box, so testing
    // against the <=50 kept boxes (2 slots/lane across 32 lanes) is exact.
    if (tid < 32) {
        const int lane = tid;
        float k0x1 = 0.f, k0y1 = 0.f, k0x2 = 0.f, k0y2 = 0.f, ka0 = 0.f;
        float k1x1 = 0.f, k1y1 = 0.f, k1x2 = 0.f, k1y2 = 0.f, ka1 = 0.f;
        int nk = 0;
        for (int i = 0; i < SORT_N; ++i) {
            float eff = skey[i];
            if (eff == NEG_INF) break;          // rest are filtered-out boxes
            int bi = sval[i];
            float cx1 = sbx[bi * 4 + 0], cy1 = sbx[bi * 4 + 1];
            float cx2 = sbx[bi * 4 + 2], cy2 = sbx[bi * 4 + 3];
            float ca  = (cx2 - cx1) * (cy2 - cy1);

            bool sup = false;
            if (lane < nk) {                    // slot 0: kept idx 0..31
                float xx1 = fmaxf(cx1, k0x1), yy1 = fmaxf(cy1, k0y1);
                float xx2 = fminf(cx2, k0x2), yy2 = fminf(cy2, k0y2);
                float inter = fmaxf(xx2 - xx1, 0.f) * fmaxf(yy2 - yy1, 0.f);
                sup = inter > NMS_THR * (ca + ka0 - inter + 1e-9f);
            }
            if (32 + lane < nk) {               // slot 1: kept idx 32..63
                float xx1 = fmaxf(cx1, k1x1), yy1 = fmaxf(cy1, k1y1);
                float xx2 = fminf(cx2, k1x2), yy2 = fminf(cy2, k1y2);
                float inter = fmaxf(xx2 - xx1, 0.f) * fmaxf(yy2 - yy1, 0.f);
                sup = sup || (inter > NMS_THR * (ca + ka1 - inter + 1e-9f));
            }
            unsigned m = __builtin_amdgcn_ballot_w32(sup);
            if (m == 0u) {                      // keep candidate
                if (lane == (nk & 31)) {
                    if (nk < 32) { k0x1 = cx1; k0y1 = cy1; k0x2 = cx2; k0y2 = cy2; ka0 = ca; }
                    else         { k1x1 = cx1; k1y1 = cy1; k1x2 = cx2; k1y2 = cy2; ka1 = ca; }
                }
                if (lane == 0) {
                    skept[nk * 4 + 0] = cx1; skept[nk * 4 + 1] = cy1;
                    skept[nk * 4 + 2] = cx2; skept[nk * 4 + 3] = cy2;
                }
                ++nk;
                if (nk == MAX_FINAL) break;
            }
        }
        if (lane == 0) s_nk = nk;
    }
    __syncthreads();

    // ---- Emit [b, x1,y1,x2,y2] rows, zero-padded past n_sel ----
    const int n_sel = min(s_nk, counts[b]);
    if (tid < MAX_FINAL) {
        const int base = (b * MAX_FINAL + tid) * 5;
        const bool v = tid < n_sel;
        out[base + 0] = (float)b;
        out[base + 1] = v ? skept[tid * 4 + 0] : 0.f;
        out[base + 2] = v ? skept[tid * 4 + 1] : 0.f;
        out[base + 3] = v ? skept[tid * 4 + 2] : 0.f;
        out[base + 4] = v ? skept[tid * 4 + 3] : 0.f;
    }
    // Tuple output #2: num_of_final_box passthrough (bit-copy int32)
    if (tid == MAX_FINAL) {
        ((int*)out)[B * MAX_FINAL * 5 + b] = counts[b];
    }
}

extern "C" void kernel_launch(void* const* d_in, const int* in_sizes, int n_in,
                              void* d_out, int out_size, void* d_ws, size_t ws_size,
                              hipStream_t stream) {
    (void)n_in; (void)out_size; (void)d_ws; (void)ws_size;
    // setup_inputs order: batch_size, im_info, batch_boxes_init, batch_scores, num_of_final_box
    const float* im_info = (const float*)d_in[1];
    const float* boxes   = (const float*)d_in[2];
    const float* scores  = (const float*)d_in[3];
    const int*   counts  = (const int*)d_in[4];
    const int B = in_sizes[4];
    const int N = in_sizes[3] / B;   // 2000; SORT_N (2048) must stay >= N
    roi_nms_kernel<<<B, BLOCK, 0, stream>>>(im_info, boxes, scores, counts,
                                            (float*)d_out, B, N);
}